// MPModule_50964081935365
// MI455X (gfx1250) — compile-verified
//
#include <hip/hip_runtime.h>

namespace {

constexpr int   B = 4, K = 21, H = 224, W = 224, NDIR = 4, NITER = 2;
constexpr float RHO  = 0.5f;
constexpr int   MSTR = B * H * W * K;          // per-direction msg stride (elements)
constexpr float FINF = __builtin_huge_valf();

// Broadcast one lane's float to all lanes via v_readlane (wave32).
static __device__ __forceinline__ float bcast(float v, int srcLane) {
  return __builtin_bit_cast(
      float, __builtin_amdgcn_readlane(__builtin_bit_cast(int, v), srcLane));
}

// One wave32 per line. msgs layout: [dir][b][y][x][k], unary layout: [b][k][y][x].
// LA/LB/LO: whether msg streams A (DIR^2), B (DIR^3), O (DIR^1) have been
// written yet this call (iteration 0 skips still-zero buffers -> no memset pass).
template <int DIR, bool LA, bool LB, bool LO>
__global__ __launch_bounds__(256) void trwp_sweep(
    const float* __restrict__ unary, const float* __restrict__ ew,
    const float* __restrict__ V, float* __restrict__ msgs) {
  const int lane = threadIdx.x & 31;
  const int wid  = threadIdx.x >> 5;
  constexpr int LINES_PER = (DIR < 2) ? H : W;   // lines per batch image
  constexpr int L         = (DIR < 2) ? W : H;   // sweep length
  const int line = blockIdx.x * 8 + wid;
  if (line >= B * LINES_PER) return;
  const int b = line / LINES_PER;
  const int l = line % LINES_PER;
  // Clamped label: lanes 21..31 exactly mirror lane 20 (same loads, same V
  // column, same broadcasts) -> identical m_raw/normalized value, so stores and
  // the 32-lane butterfly min need no lane masking at all.
  const int kl = (lane < K) ? lane : (K - 1);

  int moff, uoff, eoff, mstep, ustep, estep;
  if constexpr (DIR == 0) {          // sweep +x along row y=l
    moff  = ((b * H + l) * W + 0) * K;           mstep = K;
    uoff  = ((b * K) * H + l) * W + 0;           ustep = 1;
    eoff  = ((b * NDIR + DIR) * H + l) * W + 0;  estep = 1;
  } else if constexpr (DIR == 1) {   // sweep -x
    moff  = ((b * H + l) * W + (W - 1)) * K;          mstep = -K;
    uoff  = ((b * K) * H + l) * W + (W - 1);          ustep = -1;
    eoff  = ((b * NDIR + DIR) * H + l) * W + (W - 1); estep = -1;
  } else if constexpr (DIR == 2) {   // sweep +y along column x=l
    moff  = ((b * H + 0) * W + l) * K;           mstep = W * K;
    uoff  = ((b * K) * H + 0) * W + l;           ustep = W;
    eoff  = ((b * NDIR + DIR) * H + 0) * W + l;  estep = W;
  } else {                           // sweep -y
    moff  = ((b * H + (H - 1)) * W + l) * K;          mstep = -(W * K);
    uoff  = ((b * K) * H + (H - 1)) * W + l;          ustep = -W;
    eoff  = ((b * NDIR + DIR) * H + (H - 1)) * W + l; estep = -W;
  }
  const float* __restrict__ mA = msgs + (DIR ^ 2) * MSTR;  // other pair, dir a
  const float* __restrict__ mB = msgs + (DIR ^ 3) * MSTR;  // other pair, dir b
  const float* __restrict__ mO = msgs + (DIR ^ 1) * MSTR;  // opposite dir
  float*       __restrict__ mD = msgs + DIR * MSTR;        // direction being written
  uoff += kl * (H * W);  // unary is [b][k][y][x]; lane == output label (clamped)

  // Lane k' holds column V[:, k'] (clamped duplicate for lanes >= K).
  float Vc[K];
#pragma unroll
  for (int k = 0; k < K; ++k) Vc[k] = V[k * K + kl];

  // s = 0: message is zero (duplicate-lane store, same value same address).
  mD[moff + kl] = 0.f;

  // Software pipeline: position s-1 inputs preloaded into registers.
  float u_c = unary[uoff];
  float a_c = LA ? mA[moff + kl] : 0.f;
  float b_c = LB ? mB[moff + kl] : 0.f;
  float o_c = LO ? mO[moff + kl] : 0.f;
  eoff += estep;            // ew is consumed at position s
  float ew_c   = ew[eoff];
  float m_prev = 0.f;       // RAW (un-normalized) carry; shift-invariant min-plus

  for (int s = 1; s < L; ++s) {
    const int moff_n = moff + mstep;  // position s (always in range for s<=L-1)
    const int uoff_n = uoff + ustep;
    // Prefetch next-step inputs (read position s; used at step s+1).
    const float u_n = unary[uoff_n];
    const float a_n = LA ? mA[moff_n + kl] : 0.f;
    const float b_n = LB ? mB[moff_n + kl] : 0.f;
    const float o_n = LO ? mO[moff_n + kl] : 0.f;
    const int   eo_n = (s + 1 < L) ? (eoff + estep) : eoff;  // clamp: avoid OOB
    const float ew_n = ew[eo_n];
    {  // deep prefetch (global_prefetch_b8) keeps the L2-resident streams warm
      const int pf = (s + 8 < L) ? 8 * mstep : 0;
      if constexpr (LA) __builtin_prefetch(&mA[moff_n + pf], 0, 1);
      if constexpr (LO) __builtin_prefetch(&mO[moff_n + pf], 0, 1);
      if constexpr (!LA && !LO) __builtin_prefetch(&unary[uoff_n], 0, 1);
    }

    // base = rho*(u + sum_{d'!=d} m_d') - m_opp ; tmp = base + rho*m_prev(raw)
    const float base = RHO * (u_c + a_c + b_c + o_c) - o_c;
    const float tmp  = fmaf(RHO, m_prev, base);

    // m_raw[k'] = min_k (tmp[k] + ew*V[k,k'])  — 4 accumulators, short chain
    float a0 = FINF, a1 = FINF, a2 = FINF, a3 = FINF;
#pragma unroll
    for (int k = 0; k < 20; k += 4) {
      a0 = fminf(a0, fmaf(ew_c, Vc[k + 0], bcast(tmp, k + 0)));
      a1 = fminf(a1, fmaf(ew_c, Vc[k + 1], bcast(tmp, k + 1)));
      a2 = fminf(a2, fmaf(ew_c, Vc[k + 2], bcast(tmp, k + 2)));
      a3 = fminf(a3, fmaf(ew_c, Vc[k + 3], bcast(tmp, k + 3)));
    }
    a0 = fminf(a0, fmaf(ew_c, Vc[20], bcast(tmp, 20)));
    const float m_raw = fminf(fminf(a0, a1), fminf(a2, a3));
    m_prev = m_raw;   // recurrence continues immediately; no normalization here

    // Off-critical-path: normalize and store (butterfly min over wave32; lanes
    // >= K hold duplicates of lane 20, so no masking needed; duplicate lanes
    // store the identical value to the identical address -> guard-free).
    float mm = m_raw;
    mm = fminf(mm, __shfl_xor(mm, 16, 32));
    mm = fminf(mm, __shfl_xor(mm, 8, 32));
    mm = fminf(mm, __shfl_xor(mm, 4, 32));
    mm = fminf(mm, __shfl_xor(mm, 2, 32));
    mm = fminf(mm, __shfl_xor(mm, 1, 32));
    mD[moff_n + kl] = m_raw - mm;

    u_c = u_n; a_c = a_n; b_c = b_n; o_c = o_n; ew_c = ew_n;
    moff = moff_n; uoff = uoff_n; eoff = eo_n;
  }
}

// One block per (b, y) row. Phase 1: coalesced read of the 4 msg buffers,
// summed into an LDS tile transposed to [k][x] (pitch 233 -> stride 41 mod 64
// banks, conflict-free). Phase 2: fully coalesced unary read + out write.
constexpr int TPITCH = 233;
__global__ __launch_bounds__(256) void trwp_finalize(
    const float* __restrict__ unary, const float* __restrict__ msgs,
    float* __restrict__ out) {
  __shared__ float tile[K * TPITCH];
  const int tid = threadIdx.x;
  const int b = blockIdx.x / H;
  const int y = blockIdx.x % H;
  const int rowm = ((b * H + y) * W) * K;     // msgs row base, layout [..][x][k]
  for (int j = tid; j < W * K; j += 256) {    // j = x*K + k  (coalesced)
    const int x = j / K;
    const int k = j - x * K;
    const int mo = rowm + j;
    const float v = msgs[0 * MSTR + mo] + msgs[1 * MSTR + mo] +
                    msgs[2 * MSTR + mo] + msgs[3 * MSTR + mo];
    tile[k * TPITCH + x] = v;
  }
  __syncthreads();
  for (int j = tid; j < W * K; j += 256) {    // j = k*W + x  (coalesced)
    const int k = j / W;
    const int x = j - k * W;
    const int oi = ((b * K + k) * H + y) * W + x;   // [b][k][y][x]
    out[oi] = unary[oi] + tile[k * TPITCH + x];
  }
}

}  // namespace

extern "C" void kernel_launch(void* const* d_in, const int* in_sizes, int n_in,
                              void* d_out, int out_size, void* d_ws, size_t ws_size,
                              hipStream_t stream) {
  const float* unary = (const float*)d_in[0];  // (4,1,21,224,224)
  const float* ew    = (const float*)d_in[1];  // (4,4,224,224)
  const float* V     = (const float*)d_in[2];  // (21,21)
  float* out  = (float*)d_out;                 // (4,1,21,224,224)
  float* msgs = (float*)d_ws;                  // (4,4,224,224,21) f32, ~67.4 MB

  const int nLines = B * H;  // == B * W == 896 lines, 8 waves (lines) per block
  dim3 blk(256), grd((nLines + 7) / 8);

  // Iteration 0: skip loads of message buffers that are still unwritten
  // (their reference value is exactly zero) -> no zero-init pass needed.
  trwp_sweep<0, false, false, false><<<grd, blk, 0, stream>>>(unary, ew, V, msgs);
  trwp_sweep<1, false, false, true ><<<grd, blk, 0, stream>>>(unary, ew, V, msgs);
  trwp_sweep<2, true,  true,  false><<<grd, blk, 0, stream>>>(unary, ew, V, msgs);
  trwp_sweep<3, true,  true,  true ><<<grd, blk, 0, stream>>>(unary, ew, V, msgs);
  // Iteration 1: all message buffers live.
  trwp_sweep<0, true, true, true><<<grd, blk, 0, stream>>>(unary, ew, V, msgs);
  trwp_sweep<1, true, true, true><<<grd, blk, 0, stream>>>(unary, ew, V, msgs);
  trwp_sweep<2, true, true, true><<<grd, blk, 0, stream>>>(unary, ew, V, msgs);
  trwp_sweep<3, true, true, true><<<grd, blk, 0, stream>>>(unary, ew, V, msgs);

  trwp_finalize<<<B * H, 256, 0, stream>>>(unary, msgs, out);
}